// DiffuseBase_8538394984909
// MI455X (gfx1250) — compile-verified
//
#include <hip/hip_runtime.h>

#define DEV __device__ __forceinline__

// ---------------- problem geometry ----------------
constexpr int H = 512, W = 512, BATCH = 2;
constexpr int IMG_N = BATCH * H * W;          // 524288  (y_pred / depth)
constexpr int CV_N  = BATCH * (H - 1) * W;    // 523264
constexpr int CH_N  = BATCH * H * (W - 1);    // 523264

constexpr float LSTEP = 0.24f;
constexpr float KCOND = 0.03f;
constexpr float DEPS  = 0.1f;

// ---------------- temporal tiling -----------------
constexpr int TSTEPS = 10;               // diffusion steps fused per launch
constexpr int NLAUNCH = 500 / TSTEPS;    // 50 launches
constexpr int HALO = TSTEPS;             // halo width = steps
constexpr int TD = 63;                   // loaded tile edge (incl. halo)
constexpr int BI = TD - 2 * HALO;        // 43 interior pixels written back
constexpr int PITCH = TD + 2;            // 65 floats, LDS row pitch
constexpr int TILE_ELT = TD * PITCH;     // 4095 floats per LDS array
constexpr int NT = (H + BI - 1) / BI;    // 12 tiles per axis
static_assert(TILE_ELT * 4 * 4 <= 65536, "static LDS must fit 64KB");

// ---------------- CDNA5 async global->LDS ----------------
// gfx1250: per-lane async DMA into LDS, tracked by ASYNCcnt.
DEV void async_g2lds_b32(float* lds, const float* g) {
  unsigned l = (unsigned)(unsigned long long)lds;  // low 32 bits of generic LDS ptr = LDS offset
  asm volatile("global_load_async_to_lds_b32 %0, %1, off"
               :: "v"(l), "v"(g) : "memory");
}
DEV void wait_async0() {
  asm volatile("s_wait_asynccnt 0" ::: "memory");
}

// ---------------- float <-> order-preserving key ----------------
DEV unsigned fkey(float f) {
  unsigned u = __float_as_uint(f);
  return (u & 0x80000000u) ? ~u : (u | 0x80000000u);
}
DEV float funkey(unsigned k) {
  unsigned u = (k & 0x80000000u) ? (k & 0x7FFFFFFFu) : ~k;
  return __uint_as_float(u);
}
DEV float get_shift(const unsigned* key) {
  float m = funkey(*key);
  return (m <= DEPS) ? DEPS : 0.0f;
}
DEV unsigned umin_(unsigned a, unsigned b) { return a < b ? a : b; }

// ---------------- kernels ----------------
__global__ void k_key_init(unsigned* key) { *key = 0xFFFFFFFFu; }

__global__ __launch_bounds__(256) void k_minreduce(const float* __restrict__ x,
                                                   unsigned* __restrict__ key) {
  __shared__ unsigned s[256];
  unsigned k = 0xFFFFFFFFu;
  for (int i = blockIdx.x * 256 + threadIdx.x; i < IMG_N; i += gridDim.x * 256)
    k = umin_(k, fkey(x[i]));
  s[threadIdx.x] = k;
  __syncthreads();
  for (int off = 128; off; off >>= 1) {
    if (threadIdx.x < off) s[threadIdx.x] = umin_(s[threadIdx.x], s[threadIdx.x + off]);
    __syncthreads();
  }
  if (threadIdx.x == 0) atomicMin(key, s[0]);
}

__global__ __launch_bounds__(256) void k_init_depth(const float* __restrict__ initial,
                                                    const unsigned* __restrict__ key,
                                                    float* __restrict__ depth) {
  int i = blockIdx.x * 256 + threadIdx.x;
  if (i >= IMG_N) return;
  depth[i] = initial[i] + get_shift(key);
}

// cv[b,y,x] = g(mean_c |guide[b,c,y+1,x]-guide[b,c,y,x]|), y in [0,H-1)
__global__ __launch_bounds__(256) void k_cond_v(const float* __restrict__ guide,
                                                float* __restrict__ cvout) {
  int i = blockIdx.x * 256 + threadIdx.x;
  if (i >= CV_N) return;
  int b = i / ((H - 1) * W);
  int r = i % ((H - 1) * W);
  int y = r / W, x = r % W;
  const float* gb = guide + (long)b * 3 * H * W;
  float s = 0.f;
#pragma unroll
  for (int c = 0; c < 3; ++c) {
    const float* gc = gb + c * H * W;
    s += fabsf(gc[(y + 1) * W + x] - gc[y * W + x]);
  }
  float m = s * (1.0f / 3.0f);
  cvout[i] = 1.0f / (1.0f + (m * m) / (KCOND * KCOND));
}

// ch[b,y,x] = g(mean_c |guide[b,c,y,x+1]-guide[b,c,y,x]|), x in [0,W-1)
__global__ __launch_bounds__(256) void k_cond_h(const float* __restrict__ guide,
                                                float* __restrict__ chout) {
  int i = blockIdx.x * 256 + threadIdx.x;
  if (i >= CH_N) return;
  int b = i / (H * (W - 1));
  int r = i % (H * (W - 1));
  int y = r / (W - 1), x = r % (W - 1);
  const float* gb = guide + (long)b * 3 * H * W;
  float s = 0.f;
#pragma unroll
  for (int c = 0; c < 3; ++c) {
    const float* gc = gb + c * H * W;
    s += fabsf(gc[y * W + x + 1] - gc[y * W + x]);
  }
  float m = s * (1.0f / 3.0f);
  chout[i] = 1.0f / (1.0f + (m * m) / (KCOND * KCOND));
}

// TSTEPS fused diffusion steps, fully LDS-resident, halo-tiled.
__global__ __launch_bounds__(256) void k_diffuse(const float* __restrict__ src,
                                                 float* __restrict__ dst,
                                                 const float* __restrict__ cv,
                                                 const float* __restrict__ ch) {
  __shared__ float sA[TILE_ELT];
  __shared__ float sB[TILE_ELT];
  __shared__ float sCv[TILE_ELT];
  __shared__ float sCh[TILE_ELT];

  const int tid = threadIdx.x;
  const int b = blockIdx.z;
  const int gx0 = blockIdx.x * BI - HALO;
  const int gy0 = blockIdx.y * BI - HALO;
  const float* srcb = src + b * (H * W);
  const float* cvb  = cv  + b * ((H - 1) * W);
  const float* chb  = ch  + b * (H * (W - 1));

  // 1) zero-fill (OOB halo => zero flux, matches reference boundary handling;
  //    also guarantees no NaN garbage in the never-updated ring 0)
  for (int i = tid; i < TILE_ELT; i += 256) {
    sA[i] = 0.f; sB[i] = 0.f; sCv[i] = 0.f; sCh[i] = 0.f;
  }
  __syncthreads();

  // 2) async DMA global -> LDS for the in-range sub-rectangles
  for (int i = tid; i < TD * TD; i += 256) {
    int ly = i / TD, lx = i % TD;
    int gy = gy0 + ly, gx = gx0 + lx;
    int lidx = ly * PITCH + lx;
    if (gy >= 0 && gy < H && gx >= 0 && gx < W)
      async_g2lds_b32(&sA[lidx], srcb + gy * W + gx);
    if (gy >= 0 && gy < H - 1 && gx >= 0 && gx < W)
      async_g2lds_b32(&sCv[lidx], cvb + gy * W + gx);
    if (gy >= 0 && gy < H && gx >= 0 && gx < W - 1)
      async_g2lds_b32(&sCh[lidx], chb + gy * (W - 1) + gx);
  }
  wait_async0();
  __syncthreads();

  // 3) TSTEPS ping-pong stencil steps in LDS
  float* s0 = sA;
  float* s1 = sB;
  for (int t = 0; t < TSTEPS; ++t) {
    for (int i = tid; i < (TD - 2) * (TD - 2); i += 256) {
      int ly = 1 + i / (TD - 2);
      int lx = 1 + i % (TD - 2);
      int c = ly * PITCH + lx;
      float vC = s0[c];
      float acc = sCv[c]         * (s0[c + PITCH] - vC)
                - sCv[c - PITCH] * (vC - s0[c - PITCH])
                + sCh[c]         * (s0[c + 1] - vC)
                - sCh[c - 1]     * (vC - s0[c - 1]);
      s1[c] = vC + LSTEP * acc;
    }
    __syncthreads();
    float* tmp = s0; s0 = s1; s1 = tmp;
  }

  // 4) write back exact interior (ring >= HALO is provably uncontaminated)
  float* dstb = dst + b * (H * W);
  for (int i = tid; i < BI * BI; i += 256) {
    int iy = i / BI, ix = i % BI;
    int gy = blockIdx.y * BI + iy;
    int gx = blockIdx.x * BI + ix;
    if (gy < H && gx < W)
      dstb[gy * W + gx] = s0[(HALO + iy) * PITCH + (HALO + ix)];
  }
}

__global__ __launch_bounds__(256) void k_final(const float* __restrict__ depth,
                                               const unsigned* __restrict__ key,
                                               float* __restrict__ yp) {
  int i = blockIdx.x * 256 + threadIdx.x;
  if (i >= IMG_N) return;
  yp[i] = depth[i] - get_shift(key);
}

// ---------------- host entry ----------------
extern "C" void kernel_launch(void* const* d_in, const int* in_sizes, int n_in,
                              void* d_out, int out_size, void* d_ws, size_t ws_size,
                              hipStream_t stream) {
  (void)in_sizes; (void)n_in; (void)out_size; (void)ws_size;
  const float* guide   = (const float*)d_in[0];  // [2,3,512,512]
  const float* initial = (const float*)d_in[1];  // [2,1,512,512]

  float* out = (float*)d_out;
  float* yp = out;                 // [2,1,512,512]
  float* cv = out + IMG_N;         // [2,1,511,512]
  float* ch = out + IMG_N + CV_N;  // [2,1,512,511]

  unsigned* key = (unsigned*)d_ws;
  float* bufA = (float*)d_ws + 16;   // 64B-aligned scratch depth buffers
  float* bufB = bufA + IMG_N;

  k_key_init<<<1, 1, 0, stream>>>(key);
  k_minreduce<<<512, 256, 0, stream>>>(initial, key);
  k_init_depth<<<(IMG_N + 255) / 256, 256, 0, stream>>>(initial, key, bufA);
  k_cond_v<<<(CV_N + 255) / 256, 256, 0, stream>>>(guide, cv);
  k_cond_h<<<(CH_N + 255) / 256, 256, 0, stream>>>(guide, ch);

  dim3 grid(NT, NT, BATCH);
  for (int i = 0; i < NLAUNCH; ++i) {
    const float* s = (i & 1) ? bufB : bufA;
    float*       d = (i & 1) ? bufA : bufB;
    k_diffuse<<<grid, 256, 0, stream>>>(s, d, cv, ch);
  }
  // NLAUNCH = 50 (even) -> final result lands in bufA
  k_final<<<(IMG_N + 255) / 256, 256, 0, stream>>>(bufA, key, yp);
}